// ECELoss_17093969838131
// MI455X (gfx1250) — compile-verified
//
#include <hip/hip_runtime.h>
#include <stdint.h>

#define NBINS 15
#define NCLS 16
#define NT 256
#define NCELLS (NBINS * NCLS)            // 240
#define TILE_ROWS 128
#define TILE_ELEMS (TILE_ROWS * NCLS)    // 2048 floats = 8 KB per array per buffer

// ---- CDNA5 async copy-engine path (ASYNCcnt, ISA ch. 10.7/15.18) ------------
// global_load_async_to_lds_b128 vdst(lds byte addr), vaddr(64b), off
__device__ __forceinline__ void async_b128(unsigned lds_byte_off, const void* gaddr) {
  asm volatile("global_load_async_to_lds_b128 %0, %1, off"
               :: "v"(lds_byte_off), "v"((unsigned long long)(uintptr_t)gaddr)
               : "memory");
}
__device__ __forceinline__ void wait_async_le4() {
  asm volatile("s_wait_asynccnt 0x4" ::: "memory");
}
__device__ __forceinline__ void wait_async_0() {
  asm volatile("s_wait_asynccnt 0x0" ::: "memory");
}

// Generic pointers to LDS carry the LDS byte offset in their low 32 bits
// (flat-aperture mapping).
__device__ __forceinline__ unsigned lds_off(const void* p) {
  return (unsigned)(uintptr_t)p;
}

// Stage one 128-row tile (2048 floats per array) via 4 async B128s per thread.
__device__ __forceinline__ void async_tile(const float* gl, const float* gt,
                                           float* ll, float* lt, int tid) {
  async_b128(lds_off(ll + (size_t)tid * 4),        gl + (size_t)tid * 4);
  async_b128(lds_off(ll + (size_t)(NT + tid) * 4), gl + (size_t)(NT + tid) * 4);
  async_b128(lds_off(lt + (size_t)tid * 4),        gt + (size_t)tid * 4);
  async_b128(lds_off(lt + (size_t)(NT + tid) * 4), gt + (size_t)(NT + tid) * 4);
}

__device__ __forceinline__ float sigmoid_fast(float x) {
  // v_exp_f32 + v_rcp_f32 (trans units co-execute with VALU on CDNA5)
  return __builtin_amdgcn_rcpf(1.0f + __expf(-x));
}

// ---- Pass 1: per-block privatized histogram --------------------------------
__global__ __launch_bounds__(NT) void ece_hist_kernel(
    const float* __restrict__ logits, const float* __restrict__ targets,
    float* __restrict__ part, int B, int rowsPerBlock, int nblk) {
  // Privatized histogram: slot [bin*256 + tid] is owned exclusively by `tid`
  // (class = tid & 15). Lane-consecutive addresses -> conflict-free banks.
  __shared__ __align__(16) float    s_p [NBINS * NT];   // sum of probs
  __shared__ __align__(16) unsigned s_ct[NBINS * NT];   // (count<<16) | sum_targets
  __shared__ __align__(16) float    s_log[2][TILE_ELEMS];
  __shared__ __align__(16) float    s_tgt[2][TILE_ELEMS];

  const int tid = threadIdx.x;
#pragma unroll
  for (int b = 0; b < NBINS; ++b) { s_p[b * NT + tid] = 0.0f; s_ct[b * NT + tid] = 0u; }

  const int r0 = blockIdx.x * rowsPerBlock;
  const int r1 = min(B, r0 + rowsPerBlock);
  const int nrows = r1 - r0;
  const int ntiles = (nrows > 0) ? (nrows / TILE_ROWS) : 0;

  const float* gl0 = logits  + (size_t)r0 * NCLS;
  const float* gt0 = targets + (size_t)r0 * NCLS;

  if (ntiles > 0) async_tile(gl0, gt0, s_log[0], s_tgt[0], tid);  // prologue

  for (int t = 0; t < ntiles; ++t) {
    const int buf = t & 1;
    if (t + 1 < ntiles) {
      const size_t nb = (size_t)(t + 1) * TILE_ELEMS;
      async_tile(gl0 + nb, gt0 + nb, s_log[buf ^ 1], s_tgt[buf ^ 1], tid);
      wait_async_le4();   // next tile (4 ops) in flight; current tile complete
    } else {
      wait_async_0();
    }
    __syncthreads();
#pragma unroll
    for (int k = 0; k < TILE_ROWS / 16; ++k) {
      const int e = tid + NT * k;           // element class == tid & 15
      const float x  = s_log[buf][e];
      const float tv = s_tgt[buf][e];
      const float p  = sigmoid_fast(x);
      if (p < 1.0f) {                       // searchsorted: p==1.0 falls in no bin
        int bin = (int)(p * 15.0f);
        bin = bin > 14 ? 14 : bin;
        const int idx = bin * NT + tid;     // thread-exclusive -> no atomics needed
        s_p[idx]  += p;
        s_ct[idx] += 0x10000u + (unsigned)tv;
      }
    }
    __syncthreads();
  }

  // Remainder rows (empty when rowsPerBlock % 128 == 0): guarded direct loads.
  const int c = tid & 15, g = tid >> 4;
  for (int r = r0 + ntiles * TILE_ROWS + g; r < r1; r += 16) {
    const size_t e = (size_t)r * NCLS + c;
    const float x = logits[e], tv = targets[e];
    const float p = sigmoid_fast(x);
    if (p < 1.0f) {
      int bin = (int)(p * 15.0f);
      bin = bin > 14 ? 14 : bin;
      const int idx = bin * NT + tid;
      s_p[idx]  += p;
      s_ct[idx] += 0x10000u + (unsigned)tv;
    }
  }
  __syncthreads();

  // Reduce the 16 per-thread copies of each (class, bin) cell; write partials.
  if (tid < NCELLS) {
    const int cc = tid & 15, bb = tid >> 4;
    float sp = 0.0f;
    unsigned cnt = 0u, st = 0u;
#pragma unroll
    for (int j = 0; j < 16; ++j) {
      const int s = bb * NT + cc + (j << 4);
      sp += s_p[s];
      const unsigned v = s_ct[s];
      cnt += v >> 16;
      st  += v & 0xFFFFu;
    }
    const size_t o = (size_t)blockIdx.x * NCELLS + tid;
    const size_t stride = (size_t)nblk * NCELLS;
    part[o]              = sp;
    part[stride + o]     = (float)cnt;
    part[2 * stride + o] = (float)st;
  }
}

// ---- Pass 2: deterministic cross-block reduction + ECE ---------------------
__global__ __launch_bounds__(NT) void ece_final_kernel(
    const float* __restrict__ part, float* __restrict__ out, int nblk, int B) {
  __shared__ float rs[NT];
  __shared__ float rn[NT];
  const int i = threadIdx.x;
  float term = 0.0f, nz = 0.0f;
  if (i < NCELLS) {
    float tp = 0.0f, tc = 0.0f, tt = 0.0f;
    const size_t stride = (size_t)nblk * NCELLS;
    for (int b = 0; b < nblk; ++b) {             // fixed order -> deterministic
      const size_t o = (size_t)b * NCELLS + i;
      tp += part[o];
      tc += part[stride + o];
      tt += part[2 * stride + o];
    }
    if (tc > 0.0f) {
      term = fabsf(tp / tc - tt / tc) * (tc / (float)B);
      nz = 1.0f;
    }
  }
  rs[i] = term;
  rn[i] = nz;
  __syncthreads();
  for (int s = NT / 2; s > 0; s >>= 1) {
    if (i < s) { rs[i] += rs[i + s]; rn[i] += rn[i + s]; }
    __syncthreads();
  }
  if (i == 0) out[0] = (rn[0] > 0.0f) ? (rs[0] / rn[0]) : 0.0f;
}

extern "C" void kernel_launch(void* const* d_in, const int* in_sizes, int n_in,
                              void* d_out, int out_size, void* d_ws, size_t ws_size,
                              hipStream_t stream) {
  const float* logits  = (const float*)d_in[0];
  const float* targets = (const float*)d_in[1];
  float* out  = (float*)d_out;
  float* part = (float*)d_ws;

  const long long total = in_sizes[0];
  const int B = (int)(total / NCLS);

  int nblk = 1024;  // 1024 blocks * 240 cells * 3 floats = 2.95 MB of d_ws
  const size_t perblk = (size_t)3 * NCELLS * sizeof(float);
  if ((size_t)nblk * perblk > ws_size) {
    nblk = (int)(ws_size / perblk);
    if (nblk < 1) nblk = 1;
  }
  int rowsPerBlock = (B + nblk - 1) / nblk;
  rowsPerBlock = (rowsPerBlock + 15) & ~15;  // clean 16-thread row striding

  ece_hist_kernel<<<nblk, NT, 0, stream>>>(logits, targets, part, B, rowsPerBlock, nblk);
  ece_final_kernel<<<1, NT, 0, stream>>>(part, out, nblk, B);
}